// SCC_26491358282236
// MI455X (gfx1250) — compile-verified
//
#include <hip/hip_runtime.h>

// SCC sliding-channel conv as 8 banded GEMMs on fp32 WMMA (gfx1250),
// double-buffered LDS fed by GLOBAL_LOAD_ASYNC_TO_LDS (ASYNCcnt).
// out[b,o,n] = sum_j w[o,j] * x[b, (128*(o%8) + j) & 1023, n]

typedef float v2f __attribute__((ext_vector_type(2)));
typedef float v8f __attribute__((ext_vector_type(8)));
typedef int   v2i __attribute__((ext_vector_type(2)));

#define C_IN   1024
#define HW     3136      // 56*56
#define NBATCH 16
#define UNIT   256
#define KC     32        // K chunk staged in LDS
#define NWG    112       // N columns per workgroup (7 wmma tiles)
#define NT     7
#define MROWS  128       // rows per group
#define ASTR   34        // K-inner row stride (even -> b64-aligned, conflict-free)

#if __has_builtin(__builtin_amdgcn_global_load_async_to_lds_b32) && \
    __has_builtin(__builtin_amdgcn_global_load_async_to_lds_b64)
#define USE_ASYNC 1
#else
#define USE_ASYNC 0
#endif

// Typed pointers in explicit address spaces (AS1 = global, AS3 = LDS).
typedef __attribute__((address_space(1))) int*  g1p;
typedef __attribute__((address_space(1))) v2i*  g2p;
typedef __attribute__((address_space(3))) int*  l1p;
typedef __attribute__((address_space(3))) v2i*  l2p;

__device__ __forceinline__ g1p glb1(const void* p) { return (g1p)(unsigned long long)p; }
__device__ __forceinline__ g2p glb2(const void* p) { return (g2p)(unsigned long long)p; }
__device__ __forceinline__ l1p lds1(void* p) { return (l1p)(unsigned)(unsigned long long)p; }
__device__ __forceinline__ l2p lds2(void* p) { return (l2p)(unsigned)(unsigned long long)p; }

#if USE_ASYNC
#if __has_builtin(__builtin_amdgcn_s_wait_asynccnt)
#define WAIT_ASYNC(n) __builtin_amdgcn_s_wait_asynccnt(n)
#else
#define WAIT_ASYNC(n) asm volatile("s_wait_asynccnt %0" :: "i"(n) : "memory")
#endif
#endif

__global__ __launch_bounds__(256) void scc_wmma_f32_kernel(
    const float* __restrict__ x,      // [16,1024,56,56]
    const float* __restrict__ w,      // [1024,256]
    float* __restrict__ out)          // [16,1024,56,56]
{
    __shared__ __align__(16) float sA[2][MROWS][ASTR]; // weights tile [k][kk], K-inner
    __shared__ __align__(16) float sB[2][NWG][ASTR];   // x tile [n][kk], K-inner

    const int tid  = threadIdx.x;
    const int wave = tid >> 5;
    const int lane = tid & 31;
    const int l16  = lane & 15;
    const int hi   = lane >> 4;       // 0: K{0,1}, 1: K{2,3} within a quad
    const int r    = blockIdx.z;      // group = o % 8
    const int b    = blockIdx.y;      // batch
    const int n0   = blockIdx.x * NWG;
    const int mstrip = wave * 16;
    const int cband  = 128 * r;       // input band start for this group

    v8f acc[NT];
#pragma unroll
    for (int t = 0; t < NT; ++t) acc[t] = (v8f)0.0f;

    auto stage = [&](int buf, int kc) {
#if USE_ASYNC
        // A: 128 rows x 32 cols = 2048 b64 units -> 8 per thread (ASYNCcnt)
#pragma unroll
        for (int it = 0; it < 8; ++it) {
            int u  = it * 256 + tid;
            int k  = u >> 4;
            int j2 = (u & 15) * 2;
            __builtin_amdgcn_global_load_async_to_lds_b64(
                glb2(&w[(r + 8 * k) * UNIT + kc + j2]),
                lds2(&sA[buf][k][j2]), 0, 0);
        }
        // B: 32 K-rows x 112 cols = 3584 b32 units -> exactly 14 per thread.
        // Global side coalesced along n; LDS side stride-34 scatter (bank-clean).
#pragma unroll
        for (int it = 0; it < 14; ++it) {
            int u    = it * 256 + tid;
            int kidx = u / NWG;
            int n    = u - kidx * NWG;
            int c    = (cband + kc + kidx) & (C_IN - 1);
            __builtin_amdgcn_global_load_async_to_lds_b32(
                glb1(&x[((b * C_IN + c) * HW) + n0 + n]),
                lds1(&sB[buf][n][kidx]), 0, 0);
        }
#else
        // sync fallback
#pragma unroll
        for (int it = 0; it < (MROWS * KC) / 256; ++it) {
            int idx = it * 256 + tid;
            int k = idx >> 5;
            int j = idx & (KC - 1);
            sA[buf][k][j] = w[(r + 8 * k) * UNIT + kc + j];
        }
#pragma unroll
        for (int it = 0; it < (KC * NWG) / 256; ++it) {
            int idx = it * 256 + tid;
            int row = idx / NWG;
            int col = idx - row * NWG;
            int c = (cband + kc + row) & (C_IN - 1);
            sB[buf][col][row] = x[((b * C_IN + c) * HW) + n0 + col];
        }
#endif
    };

    stage(0, 0);
    int ib = 0;
    for (int kc = 0; kc < UNIT; kc += KC, ib ^= 1) {
#if USE_ASYNC
        if (kc + KC < UNIT) {
            stage(ib ^ 1, kc + KC);   // overlap next-chunk DMA with compute below
            WAIT_ASYNC(22);           // drain previous chunk (in-order completion)
        } else {
            WAIT_ASYNC(0);
        }
#else
        if (kc + KC < UNIT) stage(ib ^ 1, kc + KC);
#endif
        __syncthreads();

        // ---- compute: K quads; 1 A fragment reused across 7 N tiles;
        //      both fragments are single b64 LDS loads (no shuffle moves) ----
#pragma unroll
        for (int kk = 0; kk < KC; kk += 4) {
            const int kb = kk + 2 * hi;     // K{0,1} low lanes, K{2,3} high lanes
            const v2f a = *(const v2f*)&sA[ib][mstrip + l16][kb];
#pragma unroll
            for (int t = 0; t < NT; ++t) {
                const v2f bb = *(const v2f*)&sB[ib][t * 16 + l16][kb];
                acc[t] = __builtin_amdgcn_wmma_f32_16x16x4_f32(
                    false, a, false, bb, (short)0, acc[t], false, false);
            }
        }
        __syncthreads();
    }

    // ---- epilogue: C/D layout: VGPR v -> M = v + 8*hi, N = l16.
    //      NT stores: out is write-once, keep x resident in L2. ----
#pragma unroll
    for (int t = 0; t < NT; ++t) {
#pragma unroll
        for (int v = 0; v < 8; ++v) {
            int krow = mstrip + v + hi * 8;
            int o = r + 8 * krow;
            __builtin_nontemporal_store(acc[t][v],
                &out[((b * C_IN + o) * HW) + n0 + t * 16 + l16]);
        }
    }
}

extern "C" void kernel_launch(void* const* d_in, const int* in_sizes, int n_in,
                              void* d_out, int out_size, void* d_ws, size_t ws_size,
                              hipStream_t stream) {
    const float* x = (const float*)d_in[0];      // [16,1024,56,56] fp32
    const float* w = (const float*)d_in[1];      // [1024,256] fp32
    // d_in[2] = overlap (=128) -> stride 128 baked into the group decomposition.
    float* out = (float*)d_out;

    dim3 grid(HW / NWG, NBATCH, 8);              // (28, 16, 8)
    dim3 block(256);
    scc_wmma_f32_kernel<<<grid, block, 0, stream>>>(x, w, out);
}